// GCN_40776419508957
// MI455X (gfx1250) — compile-verified
//
#include <hip/hip_runtime.h>
#include <hip/hip_bf16.h>

typedef __attribute__((ext_vector_type(2))) float v2f;
typedef __attribute__((ext_vector_type(8))) float v8f;

#define NEG_FLT_MAX (-3.402823466e+38f)

// ---------------------------------------------------------------------------
// 1) deg[i] = 1.0 (self-loop contribution to degree)
__global__ void gcn_init_deg(float* __restrict__ deg, int n) {
    int i = blockIdx.x * blockDim.x + threadIdx.x;
    if (i < n) deg[i] = 1.0f;
}

// 2) deg[dst[e]] += 1 over E edges
__global__ void gcn_degree(const int* __restrict__ dst, float* __restrict__ deg, int e) {
    int i = blockIdx.x * blockDim.x + threadIdx.x;
    if (i < e) atomicAdd(&deg[dst[i]], 1.0f);
}

// 3) dinv = rsqrt(max(deg,1)); seed s with self-loop term dinv^2 * x
__global__ void gcn_dinv_seed(const float* __restrict__ deg, const float* __restrict__ x,
                              float* __restrict__ dinv, float* __restrict__ s, int n) {
    int i = blockIdx.x * blockDim.x + threadIdx.x;
    if (i < n) {
        float d = __frsqrt_rn(fmaxf(deg[i], 1.0f));
        dinv[i] = d;
        s[i] = d * d * x[i];
    }
}

// 4) layer-1 scalar edge scatter: s[v] += dinv[u]*dinv[v]*x[u]
__global__ void gcn_edge_scatter(const int* __restrict__ src, const int* __restrict__ dst,
                                 const float* __restrict__ val, const float* __restrict__ dinv,
                                 float* __restrict__ acc, int e) {
    int i = blockIdx.x * blockDim.x + threadIdx.x;
    if (i < e) {
        int u = src[i], v = dst[i];
        atomicAdd(&acc[v], dinv[u] * dinv[v] * val[u]);
    }
}

// 5) z[v] = sum_j relu(s[v]*W1[j] + b1[j]) * W2[j]  via V_WMMA_F32_16X16X4_F32.
//    One wave = one 16-node tile. A[m,k] = relu(s[m]*W1[k]+b1[k]) built in
//    registers per the ISA 16x4 f32 A layout; B[k,n] = W2[k] broadcast to all
//    16 columns; D column 0 = z for the tile. Accumulate 16 WMMAs over K=64.
//    Also seeds agg2 with the layer-2 self-loop term dinv^2 * z.
__global__ void gcn_z_wmma(const float* __restrict__ s,
                           const float* __restrict__ W1,
                           const float* __restrict__ b1,
                           const float* __restrict__ W2,
                           const float* __restrict__ dinv,
                           float* __restrict__ z,
                           float* __restrict__ agg2,
                           int nTiles) {
    const int lane  = threadIdx.x & 31;
    const int wave  = (blockIdx.x * blockDim.x + threadIdx.x) >> 5;
    const int nwave = (gridDim.x * blockDim.x) >> 5;
    const int m     = lane & 15;            // node row within tile
    const int koff  = (lane >> 4) << 1;     // lanes 0-15 -> K={0,1}; 16-31 -> K={2,3}

    for (int t = wave; t < nTiles; t += nwave) {   // uniform per wave: EXEC all-ones
        const float sv = s[t * 16 + m];
        v8f acc = (v8f)(0.0f);
        #pragma unroll
        for (int c = 0; c < 16; ++c) {             // K = 64 in chunks of 4
            const int k0 = c * 4 + koff;
            v2f a, b;
            a.x = fmaxf(fmaf(sv, W1[k0],     b1[k0]),     0.0f);
            a.y = fmaxf(fmaf(sv, W1[k0 + 1], b1[k0 + 1]), 0.0f);
            b.x = W2[k0];          // same value in every B column
            b.y = W2[k0 + 1];
            acc = __builtin_amdgcn_wmma_f32_16x16x4_f32(
                /*neg_a=*/false, a, /*neg_b=*/false, b,
                /*c_mod=*/(short)0, acc, /*reuse_a=*/false, /*reuse_b=*/false);
        }
        // D layout: lane 0 holds D[0..7,0] in acc[0..7]; lane 16 holds D[8..15,0]
        if ((lane & 15) == 0) {
            const int base = t * 16 + ((lane >> 4) << 3);
            #pragma unroll
            for (int r = 0; r < 8; ++r) {
                const int v2 = base + r;
                const float zv = acc[r];
                z[v2] = zv;
                const float dv = dinv[v2];
                agg2[v2] = dv * dv * zv;           // layer-2 self-loop term
            }
        }
    }
}

// 5b) scalar tail for N % 16 nodes (not taken for N=100000, kept for safety)
__global__ void gcn_z_tail(const float* __restrict__ s, const float* __restrict__ W1,
                           const float* __restrict__ b1, const float* __restrict__ W2,
                           const float* __restrict__ dinv, float* __restrict__ z,
                           float* __restrict__ agg2, int first, int n, int h) {
    int i = first + blockIdx.x * blockDim.x + threadIdx.x;
    if (i < n) {
        float sv = s[i], zv = 0.0f;
        for (int j = 0; j < h; ++j)
            zv = fmaf(fmaxf(fmaf(sv, W1[j], b1[j]), 0.0f), W2[j], zv);
        z[i] = zv;
        float dv = dinv[i];
        agg2[i] = dv * dv * zv;
    }
}

// 7a) per-block max of agg2 -> partials
__global__ void gcn_max_partial(const float* __restrict__ a, float* __restrict__ partials, int n) {
    __shared__ float smem[256];
    float local = NEG_FLT_MAX;
    for (int i = blockIdx.x * blockDim.x + threadIdx.x; i < n; i += gridDim.x * blockDim.x)
        local = fmaxf(local, a[i]);
    smem[threadIdx.x] = local;
    __syncthreads();
    for (int off = blockDim.x >> 1; off > 0; off >>= 1) {
        if (threadIdx.x < (unsigned)off)
            smem[threadIdx.x] = fmaxf(smem[threadIdx.x], smem[threadIdx.x + off]);
        __syncthreads();
    }
    if (threadIdx.x == 0) partials[blockIdx.x] = smem[0];
}

// 7b) reduce partials, add b2, write [1,1] output
__global__ void gcn_max_final(const float* __restrict__ partials, const float* __restrict__ b2,
                              float* __restrict__ out, int np) {
    __shared__ float smem[256];
    float local = (threadIdx.x < (unsigned)np) ? partials[threadIdx.x] : NEG_FLT_MAX;
    smem[threadIdx.x] = local;
    __syncthreads();
    for (int off = blockDim.x >> 1; off > 0; off >>= 1) {
        if (threadIdx.x < (unsigned)off)
            smem[threadIdx.x] = fmaxf(smem[threadIdx.x], smem[threadIdx.x + off]);
        __syncthreads();
    }
    if (threadIdx.x == 0) out[0] = smem[0] + b2[0];
}

extern "C" void kernel_launch(void* const* d_in, const int* in_sizes, int n_in,
                              void* d_out, int out_size, void* d_ws, size_t ws_size,
                              hipStream_t stream) {
    const float* x  = (const float*)d_in[0];   // [N,1]
    const float* W1 = (const float*)d_in[1];   // [1,H]
    const float* b1 = (const float*)d_in[2];   // [H]
    const float* W2 = (const float*)d_in[3];   // [H,1]
    const float* b2 = (const float*)d_in[4];   // [1]
    const int*   ei = (const int*)d_in[5];     // [2,E] row-major

    const int N = in_sizes[0];
    const int H = in_sizes[1];
    const int E = in_sizes[5] / 2;
    const int* src = ei;
    const int* dst = ei + E;

    // Workspace carve (floats): deg | dinv | s | z | agg2 | partials[256]
    float* deg      = (float*)d_ws;
    float* dinv     = deg  + N;
    float* s        = dinv + N;
    float* z        = s    + N;
    float* agg2     = z    + N;
    float* partials = agg2 + N;

    float* out = (float*)d_out;

    const int TB = 256;
    const int nodeBlocks = (N + TB - 1) / TB;
    const int edgeBlocks = (E + TB - 1) / TB;

    // 1) deg = 1 (self-loops)
    gcn_init_deg<<<nodeBlocks, TB, 0, stream>>>(deg, N);
    // 2) deg[dst] += 1
    gcn_degree<<<edgeBlocks, TB, 0, stream>>>(dst, deg, E);
    // 3) dinv = rsqrt(deg); s seeded with self-loop term
    gcn_dinv_seed<<<nodeBlocks, TB, 0, stream>>>(deg, x, dinv, s, N);
    // 4) layer-1 edge scatter into s
    gcn_edge_scatter<<<edgeBlocks, TB, 0, stream>>>(src, dst, x, dinv, s, E);
    // 5) dense 64-wide dot via f32 WMMA; seeds agg2 self-loop term
    const int nTiles = N / 16;
    if (nTiles > 0) {
        const int wavesPerBlock = TB / 32;
        const int wmmaBlocks = (nTiles + wavesPerBlock - 1) / wavesPerBlock;
        gcn_z_wmma<<<wmmaBlocks, TB, 0, stream>>>(s, W1, b1, W2, dinv, z, agg2, nTiles);
    }
    const int rem = N - nTiles * 16;
    if (rem > 0)
        gcn_z_tail<<<(rem + TB - 1) / TB, TB, 0, stream>>>(s, W1, b1, W2, dinv, z, agg2,
                                                           nTiles * 16, N, H);
    // 6) layer-2 edge scatter into agg2
    gcn_edge_scatter<<<edgeBlocks, TB, 0, stream>>>(src, dst, z, dinv, agg2, E);
    // 7) global max + b2
    gcn_max_partial<<<256, TB, 0, stream>>>(agg2, partials, N);
    gcn_max_final<<<1, TB, 0, stream>>>(partials, b2, out, 256);
}